// ChunkedFrameEncoder_40785009443582
// MI455X (gfx1250) — compile-verified
//
#include <hip/hip_runtime.h>
#include <hip/hip_bf16.h>

// ---- shapes (fixed by the reference) ----
#define TOTAL_K     131072
#define CHUNK_LEN   64
#define F_DIM       8
#define NUM_FRAMES  16384
#define ENC_DIM     256
#define D_MODEL     512
#define N_SCALARS   3
#define N_TICKERS   4096
#define YP          520   // padded LDS pitch for ytile (bank-conflict-free stats)

typedef float v2f __attribute__((ext_vector_type(2)));
typedef float v8f __attribute__((ext_vector_type(8)));

__device__ __forceinline__ v8f wmma_f32_4(v2f a, v2f b, v8f c) {
  // V_WMMA_F32_16X16X4_F32 : D = A(16x4) * B(4x16) + C(16x16), all f32
  return __builtin_amdgcn_wmma_f32_16x16x4_f32(
      /*neg_a=*/false, a, /*neg_b=*/false, b,
      /*c_mod=*/(short)0, c, /*reuse_a=*/false, /*reuse_b=*/false);
}

// Branchless exact-form GELU: 0.5*x*(1+erf(x/sqrt(2))) with erf via
// Abramowitz-Stegun 7.1.26 (|abs err| < 1.5e-7). Uses raw v_rcp_f32 and
// v_exp_f32 — ~15 VALU ops, no EXEC divergence (libm erff is branchy + slow).
__device__ __forceinline__ float gelu_exact(float x) {
  float z  = x * 0.70710678118654752f;
  float az = fabsf(z);
  float t  = __builtin_amdgcn_rcpf(fmaf(0.3275911f, az, 1.0f));
  float p  = fmaf(t, 1.061405429f, -1.453152027f);
  p        = fmaf(t, p,  1.421413741f);
  p        = fmaf(t, p, -0.284496736f);
  p        = fmaf(t, p,  0.254829592f);
  p       *= t;
  float e  = __expf(-az * az);
  float erf_abs = fmaf(-p, e, 1.0f);
  float erf     = copysignf(erf_abs, z);
  return 0.5f * x * (1.0f + erf);
}

// ---------------------------------------------------------------------------
// Kernel 0: zero accumulators in workspace (needed every call: atomics below)
// ---------------------------------------------------------------------------
__global__ void k_zero(float* __restrict__ p, int n) {
  int i = blockIdx.x * blockDim.x + threadIdx.x;
  if (i < n) p[i] = 0.0f;
}

// ---------------------------------------------------------------------------
// Kernel 1: per-chunk mean -> 8x256 GEMM (WMMA f32 16x16x4) -> gelu ->
//           + ticker_emb gather -> weighted atomic segment-sum into frames.
// One wave handles a 16-chunk M-tile and all 16 N-tiles (256 cols).
// grid = 1024 blocks x 256 threads (8 waves/block, 8192 M-tiles total)
// ---------------------------------------------------------------------------
__global__ void __launch_bounds__(256)
k_encode(const float* __restrict__ chunks,
         const int*   __restrict__ frame_id,
         const float* __restrict__ weights,
         const int*   __restrict__ ticker_ids,
         const float* __restrict__ W_enc,      // (8, 256) row-major
         const float* __restrict__ b_enc,      // (256)
         const float* __restrict__ ticker_emb, // (4096, 256)
         float* __restrict__ sum_emb,          // (16384, 256) accumulator
         float* __restrict__ sum_w) {          // (16384)
  const int lane  = threadIdx.x & 31;
  const int wave  = threadIdx.x >> 5;
  const int tileM = blockIdx.x * 8 + wave;     // 16-chunk tile
  const int c0    = tileM * 16;
  const int lm    = lane & 15;
  const int half  = lane >> 4;                 // 0: lanes 0-15, 1: lanes 16-31

  // ---- mean over the 64x8 window; two lanes split one chunk (32 rows each)
  const int c = c0 + lm;
  const float4* base =
      (const float4*)(chunks + (size_t)c * (CHUNK_LEN * F_DIM) + (size_t)half * 32 * F_DIM);
  __builtin_prefetch(base, 0, 0);              // global_prefetch_b8
  float p[8] = {0.f,0.f,0.f,0.f,0.f,0.f,0.f,0.f};
#pragma unroll 8
  for (int r = 0; r < 32; ++r) {
    float4 q0 = base[r * 2 + 0];
    float4 q1 = base[r * 2 + 1];
    p[0] += q0.x; p[1] += q0.y; p[2] += q0.z; p[3] += q0.w;
    p[4] += q1.x; p[5] += q1.y; p[6] += q1.z; p[7] += q1.w;
  }
  float m[8];
#pragma unroll
  for (int f = 0; f < 8; ++f)
    m[f] = (p[f] + __shfl_xor(p[f], 16, 32)) * (1.0f / 64.0f);

  // ---- A matrices, ISA 16x4 f32 layout (VGPR0: K=0|K=2, VGPR1: K=1|K=3)
  v2f A0, A1;
  A0.x = m[2 * half + 0];  A0.y = m[2 * half + 1];   // K 0..3
  A1.x = m[2 * half + 4];  A1.y = m[2 * half + 5];   // K 4..7

  // ---- per-row (D-tile rows this lane touches) gathers
  int   fid[8], tid[8];
  float wg[8];
#pragma unroll
  for (int r = 0; r < 8; ++r) {
    int cr = c0 + half * 8 + r;
    fid[r] = frame_id[cr];
    tid[r] = ticker_ids[fid[r]];
    wg[r]  = weights[cr];
  }
  // one atomic per chunk for sum_w
  if (half == 0) {
    int cc = c0 + lm;
    atomicAdd(&sum_w[frame_id[cc]], weights[cc]);
  }

  const v8f vzero = {0.f,0.f,0.f,0.f,0.f,0.f,0.f,0.f};
#pragma unroll 1
  for (int nt = 0; nt < 16; ++nt) {
    const int col = nt * 16 + lm;
    v2f B0, B1;  // mirror of A layout over K
    B0.x = W_enc[(2 * half + 0) * ENC_DIM + col];
    B0.y = W_enc[(2 * half + 1) * ENC_DIM + col];
    B1.x = W_enc[(2 * half + 4) * ENC_DIM + col];
    B1.y = W_enc[(2 * half + 5) * ENC_DIM + col];
    v8f acc = wmma_f32_4(A0, B0, vzero);
    acc     = wmma_f32_4(A1, B1, acc);

    const float be = b_enc[col];
#pragma unroll
    for (int r = 0; r < 8; ++r) {
      float h   = acc[r] + be;
      float g   = gelu_exact(h);
      float val = (g + ticker_emb[(size_t)tid[r] * ENC_DIM + col]) * wg[r];
      atomicAdd(&sum_emb[(size_t)fid[r] * ENC_DIM + col], val);
    }
  }
}

// ---------------------------------------------------------------------------
// Kernel 2: pooled(256)+scalar tail(3) @ W_proj(259x512) via WMMA f32,
//           then LayerNorm + GELU. block = 16 frames, 128 threads (4 waves),
//           wave w owns output cols [w*128, w*128+128).
// grid = 1024 blocks
// ---------------------------------------------------------------------------
__global__ void __launch_bounds__(128)
k_project(const float* __restrict__ sum_emb,
          const float* __restrict__ sum_w,
          const float* __restrict__ frame_scalars, // (16384, 3)
          const float* __restrict__ W_proj,        // (259, 512) row-major
          const float* __restrict__ b_proj,        // (512)
          const float* __restrict__ ln_gamma,      // (512)
          const float* __restrict__ ln_beta,       // (512)
          float* __restrict__ out) {               // (16384, 512)
  __shared__ float pool[16 * ENC_DIM];   // pooled embeddings, staged once
  __shared__ float ytile[16 * YP];       // pre-LN projection result (padded)
  __shared__ float smu[16], srs[16];

  const int tidx = threadIdx.x;
  const int lane = tidx & 31;
  const int wave = tidx >> 5;   // 0..3
  const int lm   = lane & 15;
  const int half = lane >> 4;
  const int fb   = blockIdx.x * 16;

  // stage pooled = sum_emb / (sum_w + 1e-8) into LDS
  for (int i = tidx; i < 16 * ENC_DIM; i += 128) {
    int row = i >> 8;
    float rw = 1.0f / (sum_w[fb + row] + 1e-8f);
    pool[i] = sum_emb[(size_t)(fb + row) * ENC_DIM + (i & 255)] * rw;
  }
  __syncthreads();

  const v8f vzero = {0.f,0.f,0.f,0.f,0.f,0.f,0.f,0.f};
  v8f acc[8];
#pragma unroll
  for (int t = 0; t < 8; ++t) acc[t] = vzero;

#pragma unroll 1
  for (int kk = 0; kk < 64; ++kk) {           // K = 256 over pooled dims
    const int kbase = 4 * kk + 2 * half;
    v2f a;                                    // A from LDS (ds_load)
    a.x = pool[lm * ENC_DIM + kbase + 0];
    a.y = pool[lm * ENC_DIM + kbase + 1];
#pragma unroll
    for (int t = 0; t < 8; ++t) {
      const int col = (wave * 8 + t) * 16 + lm;
      v2f b;
      b.x = W_proj[(size_t)(kbase + 0) * D_MODEL + col];
      b.y = W_proj[(size_t)(kbase + 1) * D_MODEL + col];
      acc[t] = wmma_f32_4(a, b, acc[t]);
    }
  }

  // scalar tail: rows 256..258 of W_proj with log1p(frame_scalars)
  float l0[8], l1[8], l2[8];
#pragma unroll
  for (int r = 0; r < 8; ++r) {
    int f = fb + half * 8 + r;
    l0[r] = log1pf(frame_scalars[f * N_SCALARS + 0]);
    l1[r] = log1pf(frame_scalars[f * N_SCALARS + 1]);
    l2[r] = log1pf(frame_scalars[f * N_SCALARS + 2]);
  }
#pragma unroll
  for (int t = 0; t < 8; ++t) {
    const int col = (wave * 8 + t) * 16 + lm;
    const float w0 = W_proj[(size_t)(ENC_DIM + 0) * D_MODEL + col];
    const float w1 = W_proj[(size_t)(ENC_DIM + 1) * D_MODEL + col];
    const float w2 = W_proj[(size_t)(ENC_DIM + 2) * D_MODEL + col];
    const float bp = b_proj[col];
#pragma unroll
    for (int r = 0; r < 8; ++r) {
      float v = acc[t][r] + bp + l0[r] * w0 + l1[r] * w1 + l2[r] * w2;
      ytile[(half * 8 + r) * YP + col] = v;
    }
  }
  __syncthreads();

  // LayerNorm stats: 8 threads per frame row, shfl-tree reduction.
  // (row groups are 8-lane aligned inside a wave; YP=520 staggers banks)
  {
    const int row = tidx >> 3;   // 0..15
    const int seg = tidx & 7;    // 0..7
    float s = 0.0f, ss = 0.0f;
#pragma unroll 16
    for (int j = 0; j < 64; ++j) {
      float v = ytile[row * YP + seg + j * 8];
      s  += v;
      ss += v * v;
    }
#pragma unroll
    for (int off = 1; off < 8; off <<= 1) {
      s  += __shfl_xor(s,  off, 32);
      ss += __shfl_xor(ss, off, 32);
    }
    if (seg == 0) {
      float mu  = s * (1.0f / (float)D_MODEL);
      float var = fmaf(-mu, mu, ss * (1.0f / (float)D_MODEL));
      smu[row] = mu;
      srs[row] = rsqrtf(fmaxf(var, 0.0f) + 1e-5f);
    }
  }
  __syncthreads();

  for (int i = tidx; i < 16 * D_MODEL; i += 128) {
    int row = i >> 9, col = i & 511;
    float x = (ytile[row * YP + col] - smu[row]) * srs[row] * ln_gamma[col] + ln_beta[col];
    out[(size_t)(fb + row) * D_MODEL + col] = gelu_exact(x);
  }
}

// ---------------------------------------------------------------------------
extern "C" void kernel_launch(void* const* d_in, const int* in_sizes, int n_in,
                              void* d_out, int out_size, void* d_ws, size_t ws_size,
                              hipStream_t stream) {
  const float* chunks        = (const float*)d_in[0];
  const int*   frame_id      = (const int*)  d_in[1];
  const float* weights       = (const float*)d_in[2];
  const float* frame_scalars = (const float*)d_in[3];
  // d_in[4] = num_frames (scalar, compile-time constant here)
  const int*   ticker_ids    = (const int*)  d_in[5];
  const float* W_enc         = (const float*)d_in[6];
  const float* b_enc         = (const float*)d_in[7];
  const float* ticker_emb    = (const float*)d_in[8];
  const float* W_proj        = (const float*)d_in[9];
  const float* b_proj        = (const float*)d_in[10];
  const float* ln_gamma      = (const float*)d_in[11];
  const float* ln_beta       = (const float*)d_in[12];
  float* out = (float*)d_out;

  float* sum_emb = (float*)d_ws;                           // 16384*256 f32
  float* sum_w   = sum_emb + (size_t)NUM_FRAMES * ENC_DIM; // 16384 f32

  const int nz = NUM_FRAMES * ENC_DIM + NUM_FRAMES;
  k_zero<<<(nz + 255) / 256, 256, 0, stream>>>(sum_emb, nz);

  // 8192 M-tiles of 16 chunks, 8 waves per block
  k_encode<<<TOTAL_K / 16 / 8, 256, 0, stream>>>(
      chunks, frame_id, weights, ticker_ids, W_enc, b_enc, ticker_emb,
      sum_emb, sum_w);

  // 16 frames per block
  k_project<<<NUM_FRAMES / 16, 128, 0, stream>>>(
      sum_emb, sum_w, frame_scalars, W_proj, b_proj, ln_gamma, ln_beta, out);
}